// GraphConvolution_16509854286319
// MI455X (gfx1250) — compile-verified
//
#include <hip/hip_runtime.h>

typedef __attribute__((ext_vector_type(2))) float v2f;
typedef __attribute__((ext_vector_type(8))) float v8f;

#define F 128  // F_IN == F_OUT == 128

// -----------------------------------------------------------------------------
// Kernel 1: support = x @ w  via V_WMMA_F32_16X16X4_F32 (fp32 WMMA, memory-bound
// GEMM so fp32 costs nothing vs fp16 and preserves reference precision).
// Block = 256 threads = 8 waves. Block computes rows [16*bid, 16*bid+16),
// wave `wid` computes columns [16*wid, 16*wid+16). K=128 in 32 steps of 4.
// -----------------------------------------------------------------------------
__global__ __launch_bounds__(256) void gemm_xw_wmma(const float* __restrict__ x,
                                                    const float* __restrict__ w,
                                                    float* __restrict__ support,
                                                    int n_rows) {
    __shared__ float wlds[F * F];  // 64 KB: full 128x128 f32 weight tile

    const int tid = threadIdx.x;
    // Cooperative load of w into LDS (row-major [k][n]); 16 float4 per thread.
    {
        const float4* wsrc = (const float4*)w;
        float4* wdst = (float4*)wlds;
#pragma unroll
        for (int i = 0; i < (F * F / 4) / 256; ++i)
            wdst[tid + i * 256] = wsrc[tid + i * 256];
    }
    __syncthreads();

    const int wid = tid >> 5;
    const int lane = tid & 31;
    const int nbase = wid * 16;
    const int rowbase = blockIdx.x * 16;
    if (rowbase >= n_rows) return;

    const int m = lane & 15;        // row (A) / column (B,C) within tile
    const int khalf = lane >> 4;    // 0: K=0,1  1: K=2,3  (32-bit A layout)

    v8f c = {};
    const float* xrow = x + (size_t)(rowbase + m) * F;

#pragma unroll 4
    for (int kb = 0; kb < F / 4; ++kb) {
        const int k0 = kb * 4 + khalf * 2;
        v2f a, b;
        // A 16x4 f32: lane<16 holds (M=lane, K=0..1), lane>=16 holds (M, K=2..3)
        a.x = xrow[k0];
        a.y = xrow[k0 + 1];
        // B 4x16 f32: same K striping, N = lane&15, from LDS
        b.x = wlds[(size_t)k0 * F + nbase + m];
        b.y = wlds[(size_t)(k0 + 1) * F + nbase + m];
        c = __builtin_amdgcn_wmma_f32_16x16x4_f32(false, a, false, b,
                                                  (short)0, c, false, false);
    }

    // C/D layout: VGPR r -> (M=r, N=lane) for lanes 0-15, (M=r+8, N=lane-16)
#pragma unroll
    for (int r = 0; r < 8; ++r) {
        const int M = r + khalf * 8;
        support[(size_t)(rowbase + M) * F + nbase + m] = c[r];
    }
}

// -----------------------------------------------------------------------------
// Kernel 2: out[dst] = sum over edges(dst) of val * support[src].
// edge_dst is sorted -> one wave per destination node, binary-search the edge
// range (wave-uniform, scalarizable). Lane l owns features [4l, 4l+4) as a
// float4 -> each gather is one coalesced 512B row read (L2-resident support).
// Deterministic: fixed per-node summation order, no float atomics.
// -----------------------------------------------------------------------------
__global__ __launch_bounds__(256) void spmm_segment(const float* __restrict__ support,
                                                    const float* __restrict__ eval,
                                                    const int* __restrict__ esrc,
                                                    const int* __restrict__ edst,
                                                    float* __restrict__ out,
                                                    int n_nodes, int n_edges) {
    const int node = blockIdx.x * (blockDim.x >> 5) + (threadIdx.x >> 5);
    if (node >= n_nodes) return;
    const int lane = threadIdx.x & 31;

    // lower_bound(edst, node)
    int lo = 0, hi = n_edges;
    while (lo < hi) {
        int mid = (lo + hi) >> 1;
        if (edst[mid] < node) lo = mid + 1; else hi = mid;
    }
    const int start = lo;
    // upper_bound(edst, node)
    hi = n_edges;
    while (lo < hi) {
        int mid = (lo + hi) >> 1;
        if (edst[mid] <= node) lo = mid + 1; else hi = mid;
    }
    const int end = lo;

    float4 acc = make_float4(0.f, 0.f, 0.f, 0.f);
    for (int e = start; e < end; ++e) {
        const int src = esrc[e];
        const float v = eval[e];
        if (e + 1 < end) {
            // Hide L2 gather latency for the next row (global_prefetch_b8)
            __builtin_prefetch(support + (size_t)esrc[e + 1] * F + lane * 4, 0, 3);
        }
        const float4 s = ((const float4*)(support + (size_t)src * F))[lane];
        acc.x += v * s.x;
        acc.y += v * s.y;
        acc.z += v * s.z;
        acc.w += v * s.w;
    }
    ((float4*)(out + (size_t)node * F))[lane] = acc;
}

extern "C" void kernel_launch(void* const* d_in, const int* in_sizes, int n_in,
                              void* d_out, int out_size, void* d_ws, size_t ws_size,
                              hipStream_t stream) {
    const float* x  = (const float*)d_in[0];
    const float* w  = (const float*)d_in[1];
    const float* ev = (const float*)d_in[2];
    const int* es   = (const int*)d_in[3];
    const int* ed   = (const int*)d_in[4];
    float* out      = (float*)d_out;

    const int n_nodes = in_sizes[0] / F;   // 100000
    const int n_edges = in_sizes[2];       // 1600000

    float* support = (float*)d_ws;         // n_nodes * F floats (51.2 MB)

    const int grid_gemm = (n_nodes + 15) / 16;  // 16 rows per block
    gemm_xw_wmma<<<grid_gemm, 256, 0, stream>>>(x, w, support, n_nodes);

    const int waves_per_block = 256 / 32;
    const int grid_spmm = (n_nodes + waves_per_block - 1) / waves_per_block;
    spmm_segment<<<grid_spmm, 256, 0, stream>>>(support, ev, es, ed, out,
                                                n_nodes, n_edges);
}